// Mlp_54468775248527
// MI455X (gfx1250) — compile-verified
//
#include <hip/hip_runtime.h>
#include <hip/hip_bf16.h>

typedef __attribute__((ext_vector_type(2))) float v2f;
typedef __attribute__((ext_vector_type(4))) float v4f;
typedef __attribute__((ext_vector_type(8))) float v8f;

#define NB    1024
#define LSEQ  128
#define DDIM  768
#define HDIM  256
#define D2    (2 * DDIM)

// ---------------------------------------------------------------------------
// 1) Ragged mean pooling, fused concat.  2048 blocks x 192 threads.
//    Each block handles one (sample, matrix) pair; thread t owns float4 #t of
//    the 768-wide row (192*4 = 768).  Fully coalesced 3 KB row reads.
//    Loads are non-temporal (each of the 805 MB is touched exactly once; no
//    point rinsing it through L2) and we prefetch 8 rows (24 KB) ahead.
// ---------------------------------------------------------------------------
__global__ void pool_concat_kernel(const float* __restrict__ uf,
                                   const float* __restrict__ hf,
                                   const int* __restrict__ ulen,
                                   const int* __restrict__ hlen,
                                   float* __restrict__ xcat) {
    int b = blockIdx.x;
    const float* src;
    float* dst;
    int len;
    if (b < NB) {
        src = uf + (size_t)b * LSEQ * DDIM;
        len = ulen[b];
        dst = xcat + (size_t)b * D2;
    } else {
        int bb = b - NB;
        src = hf + (size_t)bb * LSEQ * DDIM;
        len = hlen[bb];
        dst = xcat + (size_t)bb * D2 + DDIM;
    }
    const int t = threadIdx.x;                       // 0..191
    const v4f* s4 = (const v4f*)src;
    v4f acc = {0.f, 0.f, 0.f, 0.f};
    for (int l = 0; l < len; ++l) {
        if (l + 8 < len)                             // uniform per block
            __builtin_prefetch(&s4[(size_t)(l + 8) * (DDIM / 4) + t], 0, 0);
        v4f v = __builtin_nontemporal_load(&s4[(size_t)l * (DDIM / 4) + t]);
        acc += v;
    }
    float inv = 1.f / (float)len;
    ((v4f*)dst)[t] = acc * inv;
}

// ---------------------------------------------------------------------------
// 2) Naive transpose for weights (cheap, done once):  out[c*R + r] = in[r*C + c]
//    Thread index maps to the OUTPUT linear index -> coalesced writes.
// ---------------------------------------------------------------------------
__global__ void transpose_kernel(const float* __restrict__ in,
                                 float* __restrict__ out, int R, int C) {
    size_t idx = (size_t)blockIdx.x * blockDim.x + threadIdx.x;
    if (idx >= (size_t)R * C) return;
    int r = (int)(idx % R);
    int c = (int)(idx / R);
    out[idx] = in[(size_t)r * C + c];
}

// ---------------------------------------------------------------------------
// 3) fp32 WMMA GEMM:  C[M,N] = A[M,K] * Bt[K,N] + bias[N]
//    One wave per 32x32 output tile = 2x2 grid of 16x16 WMMA subtiles.
//    4 independent accumulators (no D->SRC2 serial chain) and 2x fragment
//    reuse: per K-step we issue 2 A-loads (b64) + 4 B-loads for 4 WMMAs.
//    A-fragment (16x4 f32): lanes 0-15 hold K=k,k+1; lanes 16-31 hold K=k+2,k+3.
//    C/D (16x16 f32, 8 VGPRs): lane<16 -> M=i, N=lane; lane>=16 -> M=i+8, N=lane-16.
// ---------------------------------------------------------------------------
__global__ void gemm_wmma_bias_kernel(const float* __restrict__ A,
                                      const float* __restrict__ Bt,
                                      const float* __restrict__ bias,
                                      float* __restrict__ C,
                                      int M, int N, int K) {
    const int lane = threadIdx.x & 31;
    const int wave = threadIdx.x >> 5;
    const int tilesN = N >> 5;                       // 32-wide macro-cols
    const int tile = blockIdx.x * (blockDim.x >> 5) + wave;
    const int tm = tile / tilesN;
    const int tn = tile - tm * tilesN;
    if (tm * 32 >= M) return;                        // wave-uniform; EXEC stays full

    const int row  = lane & 15;                      // M (for A) / N (for B/C)
    const int koff = (lane >> 4) << 1;               // 0 for lanes 0-15, 2 for 16-31

    const float* __restrict__ arow0 = A + (size_t)(tm * 32 + row) * K + koff;
    const float* __restrict__ arow1 = arow0 + (size_t)16 * K;
    const float* __restrict__ bcol0 = Bt + (size_t)koff * N + (tn * 32 + row);
    const float* __restrict__ bcol1 = bcol0 + 16;

    v8f acc00 = {};
    v8f acc01 = {};
    v8f acc10 = {};
    v8f acc11 = {};
#pragma unroll 2
    for (int k = 0; k < K; k += 4) {
        v2f a0, a1, b0, b1;
        a0.x = arow0[k];             a0.y = arow0[k + 1];
        a1.x = arow1[k];             a1.y = arow1[k + 1];
        b0.x = bcol0[(size_t)k * N]; b0.y = bcol0[(size_t)(k + 1) * N];
        b1.x = bcol1[(size_t)k * N]; b1.y = bcol1[(size_t)(k + 1) * N];
        acc00 = __builtin_amdgcn_wmma_f32_16x16x4_f32(
            false, a0, false, b0, (short)0, acc00, false, false);
        acc01 = __builtin_amdgcn_wmma_f32_16x16x4_f32(
            false, a0, false, b1, (short)0, acc01, false, false);
        acc10 = __builtin_amdgcn_wmma_f32_16x16x4_f32(
            false, a1, false, b0, (short)0, acc10, false, false);
        acc11 = __builtin_amdgcn_wmma_f32_16x16x4_f32(
            false, a1, false, b1, (short)0, acc11, false, false);
    }

    const float bv0 = bias[tn * 32 + row];
    const float bv1 = bias[tn * 32 + 16 + row];
    float* __restrict__ c00 = C + (size_t)(tm * 32) * N + tn * 32 + row;
    float* __restrict__ c01 = c00 + 16;
    float* __restrict__ c10 = c00 + (size_t)16 * N;
    float* __restrict__ c11 = c10 + 16;
#pragma unroll
    for (int i = 0; i < 8; ++i) {
        int m = (lane < 16) ? i : (i + 8);
        c00[(size_t)m * N] = acc00[i] + bv0;
        c01[(size_t)m * N] = acc01[i] + bv1;
        c10[(size_t)m * N] = acc10[i] + bv0;
        c11[(size_t)m * N] = acc11[i] + bv1;
    }
}

// ---------------------------------------------------------------------------
// 4) Deterministic column-wise sum / sum-of-squares over the batch dimension.
//    Block = 64 cols x 4 row-chunks; fixed reduction tree (no float atomics,
//    so the graph replays are bit-identical).
// ---------------------------------------------------------------------------
__global__ void colstats_kernel(const float* __restrict__ Y,
                                float* __restrict__ stats,  // [2*cols]: sum, sumsq
                                int rows, int cols) {
    __shared__ float ssum[4][64];
    __shared__ float ssq[4][64];
    const int c  = threadIdx.x & 63;
    const int rc = threadIdx.x >> 6;
    const int col = blockIdx.x * 64 + c;
    const int chunk = rows >> 2;
    const int r0 = rc * chunk;
    float s = 0.f, q = 0.f;
    for (int r = 0; r < chunk; ++r) {
        float v = Y[(size_t)(r0 + r) * cols + col];
        s += v;
        q += v * v;
    }
    ssum[rc][c] = s;
    ssq[rc][c]  = q;
    __syncthreads();
    if (rc == 0) {
        s = ssum[0][c] + ssum[1][c] + ssum[2][c] + ssum[3][c];
        q = ssq[0][c]  + ssq[1][c]  + ssq[2][c]  + ssq[3][c];
        stats[col]        = s;
        stats[cols + col] = q;
    }
}

// ---------------------------------------------------------------------------
// 5) Fused training-mode BatchNorm + ReLU, in place.
//    mean = sum/rows ; biased var = sumsq/rows - mean^2  (matches reference).
// ---------------------------------------------------------------------------
__global__ void bn_relu_kernel(float* __restrict__ Y,
                               const float* __restrict__ stats,
                               const float* __restrict__ gamma,
                               const float* __restrict__ beta,
                               int rows, int cols) {
    int idx = blockIdx.x * blockDim.x + threadIdx.x;
    if (idx >= rows * cols) return;
    int c = idx % cols;
    float inv_n = 1.f / (float)rows;
    float mu  = stats[c] * inv_n;
    float var = stats[cols + c] * inv_n - mu * mu;
    float v = (Y[idx] - mu) * rsqrtf(var + 1e-5f) * gamma[c] + beta[c];
    Y[idx] = v > 0.f ? v : 0.f;
}

// ---------------------------------------------------------------------------
// 6) fc3 (dot with a single 256-wide weight row) + sigmoid. One wave32 / row.
// ---------------------------------------------------------------------------
__global__ void fc3_sigmoid_kernel(const float* __restrict__ A,   // [rows, K]
                                   const float* __restrict__ w,   // [K]
                                   const float* __restrict__ b3,  // [1]
                                   float* __restrict__ out,       // [rows]
                                   int rows, int K) {
    int gw   = (blockIdx.x * blockDim.x + threadIdx.x) >> 5;
    int lane = threadIdx.x & 31;
    if (gw >= rows) return;
    const float* a = A + (size_t)gw * K;
    float s = 0.f;
    for (int k = lane; k < K; k += 32) s += a[k] * w[k];
#pragma unroll
    for (int off = 16; off > 0; off >>= 1) s += __shfl_down(s, off, 32);
    if (lane == 0) out[gw] = 1.f / (1.f + expf(-(s + b3[0])));
}

// ---------------------------------------------------------------------------
extern "C" void kernel_launch(void* const* d_in, const int* in_sizes, int n_in,
                              void* d_out, int out_size, void* d_ws, size_t ws_size,
                              hipStream_t stream) {
    const float* uf  = (const float*)d_in[0];
    const float* hf  = (const float*)d_in[1];
    const int*   ul  = (const int*)d_in[2];
    const int*   hl  = (const int*)d_in[3];
    const float* w1  = (const float*)d_in[4];
    const float* b1  = (const float*)d_in[5];
    const float* g1  = (const float*)d_in[6];
    const float* be1 = (const float*)d_in[7];
    const float* w2  = (const float*)d_in[8];
    const float* b2  = (const float*)d_in[9];
    const float* g2  = (const float*)d_in[10];
    const float* be2 = (const float*)d_in[11];
    const float* w3  = (const float*)d_in[12];
    const float* b3  = (const float*)d_in[13];
    float* out = (float*)d_out;

    // Workspace layout (floats, ~16 MB total)
    float* ws   = (float*)d_ws;
    float* xcat = ws;                           // 1024*1536
    float* w1t  = xcat + (size_t)NB * D2;       // 1536*768
    float* w2t  = w1t  + (size_t)D2 * DDIM;     // 768*256
    float* y1   = w2t  + (size_t)DDIM * HDIM;   // 1024*768
    float* y2   = y1   + (size_t)NB * DDIM;     // 1024*256
    float* st1  = y2   + (size_t)NB * HDIM;     // 2*768
    float* st2  = st1  + 2 * DDIM;              // 2*256

    // Weight transposes (one-time, cheap)
    transpose_kernel<<<(DDIM * D2 + 255) / 256, 256, 0, stream>>>(w1, w1t, DDIM, D2);
    transpose_kernel<<<(HDIM * DDIM + 255) / 256, 256, 0, stream>>>(w2, w2t, HDIM, DDIM);

    // Ragged mean pooling + concat (the HBM-bound stage: ~805 MB streamed)
    pool_concat_kernel<<<2 * NB, DDIM / 4, 0, stream>>>(uf, hf, ul, hl, xcat);

    // fc1: [1024,1536] x [1536,768] + bias  (fp32 WMMA, 32x32 per wave)
    {
        int tiles = (NB / 32) * (DDIM / 32);    // 32*24 = 768
        gemm_wmma_bias_kernel<<<tiles / 4, 128, 0, stream>>>(xcat, w1t, b1, y1,
                                                             NB, DDIM, D2);
    }
    colstats_kernel<<<DDIM / 64, 256, 0, stream>>>(y1, st1, NB, DDIM);
    bn_relu_kernel<<<(NB * DDIM) / 256, 256, 0, stream>>>(y1, st1, g1, be1, NB, DDIM);

    // fc2: [1024,768] x [768,256] + bias  (fp32 WMMA, 32x32 per wave)
    {
        int tiles = (NB / 32) * (HDIM / 32);    // 32*8 = 256
        gemm_wmma_bias_kernel<<<tiles / 4, 128, 0, stream>>>(y1, w2t, b2, y2,
                                                             NB, HDIM, DDIM);
    }
    colstats_kernel<<<HDIM / 64, 256, 0, stream>>>(y2, st2, NB, HDIM);
    bn_relu_kernel<<<(NB * HDIM) / 256, 256, 0, stream>>>(y2, st2, g2, be2, NB, HDIM);

    // fc3 + sigmoid -> [1024,1]
    fc3_sigmoid_kernel<<<NB / 8, 256, 0, stream>>>(y2, w3, b3, out, NB, HDIM);
}